// VanillaRNNModule_22273700397547
// MI455X (gfx1250) — compile-verified
//
#include <hip/hip_runtime.h>
#include <math.h>

typedef __attribute__((ext_vector_type(2))) float v2f;
typedef __attribute__((ext_vector_type(4))) float v4f;
typedef __attribute__((ext_vector_type(8))) float v8f;

#define B_   16
#define S_   256
#define H_   1024
#define V_   32000
#define NWG_RNN 64

// ---------------------------------------------------------------------------
// WMMA helper: D = A(16x4 f32) * B(4x16 f32) + C  (one wave)
// ---------------------------------------------------------------------------
__device__ __forceinline__ v8f wmma4(v2f a, v2f b, v8f c) {
  return __builtin_amdgcn_wmma_f32_16x16x4_f32(
      /*neg_a=*/false, a, /*neg_b=*/false, b,
      /*c_mod=*/(short)0, c, /*reuse_a=*/false, /*reuse_b=*/false);
}

// Load a 16x4 fragment from a row-major [rows x ld] f32 matrix.
// Per ISA layout: lanes 0-15 -> rows row16+lane, K = {k, k+1};
//                 lanes 16-31 -> rows row16+lane-16, K = {k+2, k+3}.
// For B = W^T (W row-major [N x K]) this same loader yields the B fragment.
__device__ __forceinline__ v2f frag_rm(const float* __restrict__ base, int ld,
                                       int row16, int k) {
  const int lane = threadIdx.x & 31;
  const float* p = base + (size_t)(row16 + (lane & 15)) * ld + k + ((lane >> 4) << 1);
  v2f r;
  r[0] = p[0];
  r[1] = p[1];
  return r;
}

// ---------------------------------------------------------------------------
// Async global->LDS copy (CDNA5): per-lane 16B, tracked by ASYNCcnt.
// lds_off = low 32 bits of the generic pointer (LDS aperture: addr[31:0]).
// ---------------------------------------------------------------------------
__device__ __forceinline__ void async_copy_b128(unsigned lds_off,
                                                const float* gaddr) {
  asm volatile("global_load_async_to_lds_b128 %0, %1, off"
               :: "v"(lds_off), "v"(gaddr) : "memory");
}
__device__ __forceinline__ void wait_async0() {
  asm volatile("s_wait_asynccnt 0x0" ::: "memory");
}

// ---------------------------------------------------------------------------
// Grid-wide barrier (persistent kernel), sense via generation counter
// ---------------------------------------------------------------------------
__device__ __forceinline__ void grid_sync(unsigned* cnt, unsigned* gen, unsigned nwg) {
  __threadfence();          // flush this thread's global stores (release)
  __syncthreads();
  if (threadIdx.x == 0) {
    unsigned g = atomicAdd(gen, 0u);
    if (atomicAdd(cnt, 1u) == nwg - 1u) {
      atomicExch(cnt, 0u);
      __threadfence();
      atomicAdd(gen, 1u);
    } else {
      while (atomicAdd(gen, 0u) == g) { __builtin_amdgcn_s_sleep(2); }
    }
    __threadfence();
  }
  __syncthreads();
}

// ---------------------------------------------------------------------------
// Kernel 1: embedding gather  emb[s][b][h] = embed[x[b][s]][h]
// ---------------------------------------------------------------------------
__global__ void __launch_bounds__(256) gather_kernel(
    const int* __restrict__ x, const float* __restrict__ embed,
    float* __restrict__ emb) {
  const int bid = blockIdx.x;            // = s*16 + b
  const int s = bid >> 4, b = bid & 15;
  int tok = x[b * S_ + s];
  if (tok < 0) tok = 0;
  if (tok >= V_) tok = V_ - 1;
  const v4f* src = (const v4f*)(embed + (size_t)tok * H_);
  v4f* dst = (v4f*)(emb + (size_t)bid * H_);
  dst[threadIdx.x] = src[threadIdx.x];   // 256 threads * 16B = 4 KB row
}

// ---------------------------------------------------------------------------
// Kernel 2: persistent RNN. 64 WGs x 256 thr; WG owns neuron tile j0..j0+15
// of both layers. Per layer-step: 8 waves split virtual K = 2048
// (inp@Wi^T concat h@Wh^T), partial 16x16 tiles reduced through LDS,
// then bias + tanh + store. Weights stay L2-resident across all 256 steps.
// ---------------------------------------------------------------------------
__device__ __forceinline__ void rnn_layer(
    const float* __restrict__ inp,    // [16][1024]
    const float* __restrict__ hprev,  // [16][1024]
    const float* __restrict__ Wi_l,   // [1024][1024] row-major [j][k]
    const float* __restrict__ Wh_l,
    const float* __restrict__ bi_l, const float* __restrict__ bh_l,
    int j0, float* __restrict__ hout, float* __restrict__ tout,
    float* lds /* 2048 floats */) {
  const int tid  = threadIdx.x;
  const int w    = tid >> 5;    // wave 0..7
  const int lane = tid & 31;
  const float* A = (w < 4) ? inp  : hprev;
  const float* W = (w < 4) ? Wi_l : Wh_l;
  const int kb   = (w & 3) * 256;

  v8f c = {};
  for (int k = 0; k < 256; k += 4) {
    v2f a = frag_rm(A, H_, 0, kb + k);
    v2f b = frag_rm(W, H_, j0, kb + k);   // B[k][n] = W[j0+n][k]
    c = wmma4(a, b, c);
  }
#pragma unroll
  for (int r = 0; r < 8; ++r) lds[w * 256 + r * 32 + lane] = c[r];
  __syncthreads();

  float v = 0.f;
#pragma unroll
  for (int p = 0; p < 8; ++p) v += lds[p * 256 + tid];
  const int r  = tid >> 5, ln = tid & 31;
  const int m  = r + ((ln >> 4) << 3);   // batch row 0..15
  const int n  = ln & 15;
  const int j  = j0 + n;
  v = tanhf(v + bi_l[j] + bh_l[j]);
  hout[m * H_ + j] = v;
  if (tout) tout[m * H_ + j] = v;
  __syncthreads();                        // LDS reuse guard
}

__global__ void __launch_bounds__(256) rnn_kernel(
    const float* __restrict__ emb, const float* __restrict__ Wi,
    const float* __restrict__ bi, const float* __restrict__ Wh,
    const float* __restrict__ bh, float* __restrict__ hA,
    float* __restrict__ hB, float* __restrict__ tops,
    unsigned* __restrict__ bar, float* __restrict__ hid_out) {
  __shared__ float lds[2048];
  const int j0 = blockIdx.x * 16;
  const int LW = H_ * H_;   // per-layer weight stride
  for (int t = 0; t < S_; ++t) {
    const float* rd = (t & 1) ? hB : hA;
    float*       wr = (t & 1) ? hA : hB;
    // layer 0
    rnn_layer(emb + (size_t)t * B_ * H_, rd, Wi, Wh, bi, bh, j0,
              wr, nullptr, lds);
    grid_sync(bar, bar + 1, NWG_RNN);
    // layer 1 (input = fresh h0)
    rnn_layer(wr, rd + B_ * H_, Wi + LW, Wh + LW, bi + H_, bh + H_, j0,
              wr + B_ * H_, tops + (size_t)t * B_ * H_, lds);
    grid_sync(bar, bar + 1, NWG_RNN);
  }
  // t=255 (odd) wrote hA -> final hidden lives in hA. Copy [2][16][1024].
  const int gtid = blockIdx.x * 256 + threadIdx.x;    // 0..16383
  hid_out[gtid]          = hA[gtid];
  hid_out[gtid + 16384]  = hA[gtid + 16384];
}

// ---------------------------------------------------------------------------
// Kernel 3: output projection. C[4096 x 32000] = tops[4096 x 1024] @ Wo^T + bo
// Workgroup tile 128x128; K chunked (Kc=16) with DOUBLE-BUFFERED async
// global->LDS staging (ASYNCcnt pipeline); 8 waves each own a 128x16 strip
// (8 wmma tiles). Logits streamed out with nontemporal stores so the 524 MB
// result doesn't evict the L2-resident Wo.
// ---------------------------------------------------------------------------
#define KC   16
#define LDKP 20   // padded LDS row stride: 80 B -> float4 slots stay 16B-aligned,
                  // and 20*i mod 64 distinct over 16 rows -> conflict-free frags

__device__ __forceinline__ void stage_async(
    const float* __restrict__ gA, const float* __restrict__ gB,
    float* sA, float* sB, int m0, int v0, int kc) {
  const int tid = threadIdx.x;
#pragma unroll
  for (int p = 0; p < 2; ++p) {
    const int q   = tid + p * 256;    // float4 slot 0..511 (128 rows x 4 slots)
    const int row = q >> 2;
    const int col = (q & 3) << 2;
    const float* ga = gA + (size_t)(m0 + row) * H_ + kc + col;
    const float* gb = gB + (size_t)(v0 + row) * H_ + kc + col;
    const unsigned la = (unsigned)(unsigned long long)(sA + row * LDKP + col);
    const unsigned lb = (unsigned)(unsigned long long)(sB + row * LDKP + col);
    async_copy_b128(la, ga);
    async_copy_b128(lb, gb);
  }
}

__global__ void __launch_bounds__(256) proj_kernel(
    const float* __restrict__ tops, const float* __restrict__ Wo,
    const float* __restrict__ bo, float* __restrict__ out) {
  __shared__ __align__(16) float As[2][128 * LDKP];
  __shared__ __align__(16) float Bs[2][128 * LDKP];
  const int tid = threadIdx.x;
  const int w = tid >> 5, lane = tid & 31;
  const int m0 = blockIdx.x * 128;   // 32 blocks over 4096 rows (r = s*16+b)
  const int v0 = blockIdx.y * 128;   // 250 blocks over 32000 cols

  v8f c[8];
#pragma unroll
  for (int i = 0; i < 8; ++i) c[i] = (v8f){};

  // prologue: fill buffer 0
  stage_async(tops, Wo, As[0], Bs[0], m0, v0, 0);
  wait_async0();
  __syncthreads();

  int buf = 0;
  for (int kc = 0; kc < H_; kc += KC) {
    // issue next chunk's async copies into the other buffer
    if (kc + KC < H_)
      stage_async(tops, Wo, As[buf ^ 1], Bs[buf ^ 1], m0, v0, kc + KC);

    // compute on current buffer
    const float* sA = As[buf];
    const float* sB = Bs[buf];
#pragma unroll
    for (int kk = 0; kk < KC; kk += 4) {
      v2f bfrag = frag_rm(sB, LDKP, w * 16, kk);   // this wave's 16 v-columns
#pragma unroll
      for (int mt = 0; mt < 8; ++mt) {
        v2f afrag = frag_rm(sA, LDKP, mt * 16, kk);
        c[mt] = wmma4(afrag, bfrag, c[mt]);
      }
    }

    wait_async0();       // next buffer's data landed in LDS
    __syncthreads();     // and everyone is done reading the current buffer
    buf ^= 1;
  }

  // epilogue: scatter to out[b][s][v] (+ bias), nontemporal (streaming) stores
  const int nl = lane & 15;
  const int v  = v0 + w * 16 + nl;
  const float bias = bo[v];
  const int hi = (lane >> 4) << 3;
#pragma unroll
  for (int mt = 0; mt < 8; ++mt) {
#pragma unroll
    for (int r = 0; r < 8; ++r) {
      const int row = m0 + mt * 16 + r + hi;   // = s*16 + b
      const int s = row >> 4, b = row & 15;
      __builtin_nontemporal_store(c[mt][r] + bias,
                                  &out[((size_t)(b * S_ + s)) * V_ + v]);
    }
  }
}

// ---------------------------------------------------------------------------
// Host launcher
// ---------------------------------------------------------------------------
extern "C" void kernel_launch(void* const* d_in, const int* in_sizes, int n_in,
                              void* d_out, int out_size, void* d_ws, size_t ws_size,
                              hipStream_t stream) {
  const int*   x     = (const int*)d_in[0];
  const float* embed = (const float*)d_in[1];
  const float* Wi    = (const float*)d_in[2];
  const float* bi    = (const float*)d_in[3];
  const float* Wh    = (const float*)d_in[4];
  const float* bh    = (const float*)d_in[5];
  const float* Wo    = (const float*)d_in[6];
  const float* bo    = (const float*)d_in[7];
  float* out = (float*)d_out;

  // workspace layout (bytes)
  char* ws = (char*)d_ws;
  const size_t EMB_BYTES = (size_t)S_ * B_ * H_ * 4;  // 16 MB
  float* emb  = (float*)(ws);
  float* tops = (float*)(ws + EMB_BYTES);
  float* hA   = (float*)(ws + 2 * EMB_BYTES);
  float* hB   = (float*)(ws + 2 * EMB_BYTES + 2 * B_ * H_ * 4);
  unsigned* bar = (unsigned*)(ws + 2 * EMB_BYTES + 4 * B_ * H_ * 4);

  // init h0 = 0 and the barrier state (graph-capture safe)
  hipMemsetAsync(hA, 0, 2 * B_ * H_ * 4, stream);
  hipMemsetAsync(hB, 0, 2 * B_ * H_ * 4, stream);
  hipMemsetAsync(bar, 0, 256, stream);

  gather_kernel<<<S_ * B_, 256, 0, stream>>>(x, embed, emb);
  rnn_kernel<<<NWG_RNN, 256, 0, stream>>>(emb, Wi, bi, Wh, bh, hA, hB, tops,
                                          bar, out + (size_t)B_ * S_ * V_);
  proj_kernel<<<dim3(32, 250), 256, 0, stream>>>(tops, Wo, bo, out);
}